// Encoder_89558658056522
// MI455X (gfx1250) — compile-verified
//
#include <hip/hip_runtime.h>

// Problem constants (match reference)
#define S_DIM 2048
#define N_DIM 32768
#define NFLOW 30
#define S_PER_BLOCK 8

typedef __attribute__((ext_vector_type(2))) float v2f;
typedef __attribute__((ext_vector_type(4))) float v4f;
typedef __attribute__((ext_vector_type(8))) float v8f;

__device__ __forceinline__ float softplus_f(float x) {
    // numerically stable softplus
    return fmaxf(x, 0.0f) + log1pf(expf(-fabsf(x)));
}

// ---------------------------------------------------------------------------
// Kernel 1: base-dist sample + 30 radial flow steps. One thread per sample.
// Writes Z[s][4]  (= zK reshaped to 2x2 row-major: Z[d][q] at index d*2+q)
// ---------------------------------------------------------------------------
__global__ __launch_bounds__(256) void flow_kernel(
    const float* __restrict__ eps, const float* __restrict__ mu,
    const float* __restrict__ sigma, const float* __restrict__ x0,
    const float* __restrict__ a_p, const float* __restrict__ b_p,
    float* __restrict__ Zout) {
    int s = blockIdx.x * blockDim.x + threadIdx.x;
    if (s >= S_DIM) return;
    float z[4];
#pragma unroll
    for (int d = 0; d < 4; ++d) {
        float m  = 3.0f * tanhf(mu[d]);
        float sg = tanhf(sigma[d]) + 1.0f;
        z[d] = m + sg * eps[s * 4 + d];
    }
    for (int i = 0; i < NFLOW; ++i) {
        float alpha = softplus_f(a_p[i]);
        float beta  = -alpha + softplus_f(b_p[i]);
        float diff[4];
        float r2 = 0.0f;
#pragma unroll
        for (int d = 0; d < 4; ++d) {
            diff[d] = z[d] - x0[i * 4 + d];
            r2 += diff[d] * diff[d];
        }
        float r  = sqrtf(r2);
        float bh = beta / (alpha + r);
#pragma unroll
        for (int d = 0; d < 4; ++d) z[d] += bh * diff[d];
    }
    v4f zv = {z[0], z[1], z[2], z[3]};
    *(v4f*)(Zout + s * 4) = zv;
}

// ---------------------------------------------------------------------------
// Kernel 2: MLP f = tanh(relu(relu(Y@W1+b1)@W2+b2)@W3+b3) via f32 WMMA.
// Each wave computes a 16-row tile; K=10 padded to 12 -> 3x WMMA 16x16x4.
// All padded loads use clamped (in-bounds) indices + value select, so there
// are no divergent exec-mask branches around the loads.
// ---------------------------------------------------------------------------
__device__ __forceinline__ float masked_ld(const float* __restrict__ p, int idx,
                                           int idx_clamped, bool valid) {
    float v = p[idx_clamped];
    (void)idx;
    return valid ? v : 0.0f;
}

__device__ __forceinline__ v2f load_B_chunk(const float* __restrict__ W, int ncols,
                                            int kc, int half, int lrow,
                                            int kmax, int nmax) {
    // B 4x16 layout: v0 -> K = kc + 2*half, v1 -> K = kc + 2*half + 1, N = lrow
    int K0  = kc + 2 * half;
    int K0c = min(K0, kmax - 1);
    int K1c = min(K0 + 1, kmax - 1);
    int nc  = min(lrow, nmax - 1);
    bool nok = (lrow < nmax);
    v2f b;
    b.x = masked_ld(W, 0, K0c * ncols + nc, (K0 < kmax) && nok);
    b.y = masked_ld(W, 0, K1c * ncols + nc, (K0 + 1 < kmax) && nok);
    return b;
}

__global__ __launch_bounds__(256) void mlp_kernel(
    const float* __restrict__ Y,
    const float* __restrict__ W1, const float* __restrict__ b1,
    const float* __restrict__ W2, const float* __restrict__ b2,
    const float* __restrict__ W3, const float* __restrict__ b3,
    float* __restrict__ F) {
    __shared__ float lds[8 * 256];  // 8 waves * 16x16 f32 tile
    const int lane = threadIdx.x & 31;
    const int wave = threadIdx.x >> 5;
    const int half = lane >> 4;   // 0: lanes 0-15, 1: lanes 16-31
    const int lrow = lane & 15;   // M for A-frag, N for B/C/D frags
    const int base = (blockIdx.x * 8 + wave) * 16;  // first of 16 Y rows
    float* tile = lds + wave * 256;

    // ---- layer 1: A from Y (16x10), B from W1 (10x10) ----
    v8f acc = {};
#pragma unroll
    for (int c = 0; c < 3; ++c) {
        int kc = 4 * c;
        int K0 = kc + 2 * half;
        v2f a;
        a.x = masked_ld(Y, 0, (base + lrow) * 10 + min(K0, 9), K0 < 10);
        a.y = masked_ld(Y, 0, (base + lrow) * 10 + min(K0 + 1, 9), K0 + 1 < 10);
        v2f b = load_B_chunk(W1, 10, kc, half, lrow, 10, 10);
        acc = __builtin_amdgcn_wmma_f32_16x16x4_f32(false, a, false, b,
                                                    (short)0, acc, false, false);
    }
    {
        float bias = (lrow < 10) ? b1[min(lrow, 9)] : 0.0f;
#pragma unroll
        for (int v = 0; v < 8; ++v) {
            float h = fmaxf(acc[v] + bias, 0.0f);     // relu
            tile[(v + 8 * half) * 16 + lrow] = h;     // D-layout -> row-major LDS
        }
    }

    // ---- layer 2: A from LDS tile, B from W2 ----
    v2f a2[3];
#pragma unroll
    for (int c = 0; c < 3; ++c) {   // read all A frags before overwriting tile
        int K0 = 4 * c + 2 * half;
        float x = tile[lrow * 16 + min(K0, 9)];
        float y = tile[lrow * 16 + min(K0 + 1, 9)];
        a2[c].x = (K0 < 10) ? x : 0.0f;
        a2[c].y = (K0 + 1 < 10) ? y : 0.0f;
    }
    acc = (v8f){};
#pragma unroll
    for (int c = 0; c < 3; ++c) {
        v2f b = load_B_chunk(W2, 10, 4 * c, half, lrow, 10, 10);
        acc = __builtin_amdgcn_wmma_f32_16x16x4_f32(false, a2[c], false, b,
                                                    (short)0, acc, false, false);
    }
    {
        float bias = (lrow < 10) ? b2[min(lrow, 9)] : 0.0f;
#pragma unroll
        for (int v = 0; v < 8; ++v) {
            float h = fmaxf(acc[v] + bias, 0.0f);
            tile[(v + 8 * half) * 16 + lrow] = h;
        }
    }

    // ---- layer 3: A from LDS tile, B from W3 (10x2) ----
    v2f a3[3];
#pragma unroll
    for (int c = 0; c < 3; ++c) {
        int K0 = 4 * c + 2 * half;
        float x = tile[lrow * 16 + min(K0, 9)];
        float y = tile[lrow * 16 + min(K0 + 1, 9)];
        a3[c].x = (K0 < 10) ? x : 0.0f;
        a3[c].y = (K0 + 1 < 10) ? y : 0.0f;
    }
    acc = (v8f){};
#pragma unroll
    for (int c = 0; c < 3; ++c) {
        v2f b = load_B_chunk(W3, 2, 4 * c, half, lrow, 10, 2);
        acc = __builtin_amdgcn_wmma_f32_16x16x4_f32(false, a3[c], false, b,
                                                    (short)0, acc, false, false);
    }
    if (lrow < 2) {
        float bias = b3[lrow & 1];
#pragma unroll
        for (int v = 0; v < 8; ++v) {
            int row = base + v + 8 * half;
            F[row * 2 + lrow] = tanhf(acc[v] + bias);
        }
    }
}

// ---------------------------------------------------------------------------
// Kernel 3: X_mean[s,n,q] = sum_d f[n,d] * Z[s,d,q]. Store-bandwidth bound:
// 512 MB output -> ~22 us floor at 23.3 TB/s. Each thread keeps its f float4
// in registers and loops over 8 s-values -> f L2-read traffic cut 8x; one
// 16B non-temporal store per (s, n-pair).
// ---------------------------------------------------------------------------
__global__ __launch_bounds__(256) void einsum_kernel(
    const float* __restrict__ Zbuf, const float* __restrict__ F,
    float* __restrict__ out) {
    const int t  = blockIdx.x * blockDim.x + threadIdx.x;
    const int n0 = t * 2;
    const int s0 = blockIdx.y * S_PER_BLOCK;
    // f[n0,0], f[n0,1], f[n0+1,0], f[n0+1,1] — register-resident across 8 s
    const v4f f = *(const v4f*)(F + n0 * 2);
#pragma unroll
    for (int i = 0; i < S_PER_BLOCK; ++i) {
        const int s = s0 + i;
        // Z[s]: [d][q] row-major: [0]=Z00 [1]=Z01 [2]=Z10 [3]=Z11 (L1-hot)
        const v4f Z = *(const v4f*)(Zbuf + s * 4);
        v4f o;
        o.x = f.x * Z.x + f.y * Z.z;
        o.y = f.x * Z.y + f.y * Z.w;
        o.z = f.z * Z.x + f.w * Z.z;
        o.w = f.z * Z.y + f.w * Z.w;
        // 512 MB write stream: non-temporal so it doesn't evict hot F/Z
        __builtin_nontemporal_store(
            o, (v4f*)(out + (size_t)s * (N_DIM * 2) + n0 * 2));
    }
}

extern "C" void kernel_launch(void* const* d_in, const int* in_sizes, int n_in,
                              void* d_out, int out_size, void* d_ws, size_t ws_size,
                              hipStream_t stream) {
    const float* eps   = (const float*)d_in[0];
    const float* mu    = (const float*)d_in[1];
    const float* sigma = (const float*)d_in[2];
    const float* x0    = (const float*)d_in[3];
    const float* a_p   = (const float*)d_in[4];
    const float* b_p   = (const float*)d_in[5];
    const float* Y     = (const float*)d_in[6];
    const float* W1    = (const float*)d_in[7];
    const float* b1    = (const float*)d_in[8];
    const float* W2    = (const float*)d_in[9];
    const float* b2    = (const float*)d_in[10];
    const float* W3    = (const float*)d_in[11];
    const float* b3    = (const float*)d_in[12];
    float* out = (float*)d_out;

    float* ws   = (float*)d_ws;
    float* Zbuf = ws;               // S_DIM*4  = 8192 floats
    float* Fbuf = ws + S_DIM * 4;   // N_DIM*2  = 65536 floats

    // 1) flows: 2048 samples, 1 thread each
    hipLaunchKernelGGL(flow_kernel, dim3(S_DIM / 256), dim3(256), 0, stream,
                       eps, mu, sigma, x0, a_p, b_p, Zbuf);
    // 2) MLP via WMMA: 16 rows/wave, 8 waves/block -> 128 rows/block
    hipLaunchKernelGGL(mlp_kernel, dim3(N_DIM / 128), dim3(256), 0, stream,
                       Y, W1, b1, W2, b2, W3, b3, Fbuf);
    // 3) bandwidth kernel: grid (N/512, S/8); each thread: 1 f-load, 8 NT stores
    hipLaunchKernelGGL(einsum_kernel,
                       dim3(N_DIM / 512, S_DIM / S_PER_BLOCK), dim3(256), 0, stream,
                       Zbuf, Fbuf, out);
}